// GNNEncoder_31421980737623
// MI455X (gfx1250) — compile-verified
//
#include <hip/hip_runtime.h>
#include <stdint.h>

typedef float v2f __attribute__((ext_vector_type(2)));
typedef float v8f __attribute__((ext_vector_type(8)));

#define BN_EPS 1e-5f

// ---------------- degree / normalization ----------------

__global__ void k_deg_init(float* __restrict__ deg, int n) {
    int i = blockIdx.x * blockDim.x + threadIdx.x;
    if (i < n) deg[i] = 1.0f;  // self loop contributes 1
}

__global__ void k_deg_count(const int64_t* __restrict__ dst, int E, float* __restrict__ deg) {
    int i = blockIdx.x * blockDim.x + threadIdx.x;
    int stride = gridDim.x * blockDim.x;
    for (int e = i; e < E; e += stride)
        atomicAdd(&deg[(int)dst[e]], 1.0f);
}

__global__ void k_dinv(float* __restrict__ deg, int n) {
    int i = blockIdx.x * blockDim.x + threadIdx.x;
    if (i < n) deg[i] = rsqrtf(deg[i]);
}

// ---------------- WMMA f32 GEMM: H[n][DOUT] = X[n][DIN] @ W[DOUT][DIN]^T ----------------
// Each wave owns ONE 16-column n-tile and FOUR concurrent 16-row tiles.
//  - B fragments (this n-tile, all K) are loaded once into registers: DIN/4 x v2f = 64 VGPRs.
//  - K is consumed in chunks of 4 steps: 16 batched A loads -> one wait -> 16 WMMAs
//    interleaved across 4 independent accumulator chains (hides WMMA RAW hazard).
// Fragment layouts per CDNA5 ISA:
//  A 16x4 f32: lanes 0-15 hold K=0,1 for M=lane; lanes 16-31 hold K=2,3.
//  B 4x16 f32: lane holds column N=lane&15, K offset 2*(lane>>4).
//  C/D 16x16 f32: VGPR r -> M=r (+8 for lanes 16-31); N=lane&15.

template <int DIN, int DOUT>
__global__ void k_gemm_wmma(const float* __restrict__ X, const float* __restrict__ W,
                            float* __restrict__ H, int n_nodes) {
    constexpr int KS  = DIN / 4;   // k-steps (32)
    constexpr int NTL = DOUT / 16; // n tiles (8 or 4)
    constexpr int RC  = 4;         // concurrent row tiles per wave
    constexpr int KC  = 4;         // k-steps per chunk

    const int lane  = threadIdx.x & 31;
    const int gwave = (blockIdx.x * blockDim.x + threadIdx.x) >> 5;
    const int ntile = gwave % NTL;
    const int rquad = gwave / NTL;

    const int lrow  = lane & 15;
    const int khalf = (lane >> 4) * 2;  // 0 or 2

    const int row_tile0 = rquad * RC;
    if (row_tile0 * 16 >= n_nodes) return;  // wave-uniform

    // register-resident B fragments for this n-tile
    v2f b[KS];
    {
        const float* wrow = W + (size_t)(ntile * 16 + lrow) * DIN + khalf;
#pragma unroll
        for (int ks = 0; ks < KS; ++ks) {
            b[ks][0] = wrow[ks * 4];
            b[ks][1] = wrow[ks * 4 + 1];
        }
    }

    // A row pointers; tail quads clamp to a valid row (stores are guarded)
    const float* xrow[RC];
#pragma unroll
    for (int rc = 0; rc < RC; ++rc) {
        int row0 = (row_tile0 + rc) * 16;
        int r = (row0 < n_nodes) ? (row0 + lrow) : (n_nodes - 16 + lrow);
        xrow[rc] = X + (size_t)r * DIN + khalf;
    }

    v8f acc[RC];
    v8f zero = {0.f, 0.f, 0.f, 0.f, 0.f, 0.f, 0.f, 0.f};
#pragma unroll
    for (int rc = 0; rc < RC; ++rc) acc[rc] = zero;

    for (int kc = 0; kc < KS; kc += KC) {
        v2f a[RC][KC];
#pragma unroll
        for (int rc = 0; rc < RC; ++rc)
#pragma unroll
            for (int j = 0; j < KC; ++j) {
                a[rc][j][0] = xrow[rc][(kc + j) * 4];
                a[rc][j][1] = xrow[rc][(kc + j) * 4 + 1];
            }
#pragma unroll
        for (int j = 0; j < KC; ++j)
#pragma unroll
            for (int rc = 0; rc < RC; ++rc)
                acc[rc] = __builtin_amdgcn_wmma_f32_16x16x4_f32(
                    false, a[rc][j], false, b[kc + j],
                    (short)0, acc[rc], false, false);
    }

    const int mhi = (lane >> 4) * 8;
#pragma unroll
    for (int rc = 0; rc < RC; ++rc) {
        int row0 = (row_tile0 + rc) * 16;
        if (row0 >= n_nodes) break;  // wave-uniform guard
        float* hp = H + (size_t)row0 * DOUT + ntile * 16 + lrow;
#pragma unroll
        for (int r = 0; r < 8; ++r)
            hp[(size_t)(mhi + r) * DOUT] = acc[rc][r];
    }
}

// ---------------- aggregation ----------------

// out[i][f] = H[i][f] * dinv[i]^2 (self loop) + bias[f]
template <int D>
__global__ void k_init_agg(const float* __restrict__ H, const float* __restrict__ dinv,
                           const float* __restrict__ bias, float* __restrict__ out, int total) {
    int idx = blockIdx.x * blockDim.x + threadIdx.x;
    if (idx < total) {
        int i = idx / D, f = idx % D;
        float di = dinv[i];
        out[idx] = H[idx] * di * di + bias[f];
    }
}

// One wave per edge (grid-stride); lane covers D/32 consecutive features.
template <int D>
__global__ void k_scatter(const float* __restrict__ H, const int64_t* __restrict__ src,
                          const int64_t* __restrict__ dst, const float* __restrict__ dinv,
                          float* __restrict__ out, int E) {
    constexpr int VEC = D / 32;  // 4 (D=128) or 2 (D=64)
    int gwave = (blockIdx.x * blockDim.x + threadIdx.x) >> 5;
    int lane  = threadIdx.x & 31;
    int nwaves = (gridDim.x * blockDim.x) >> 5;
    for (int e = gwave; e < E; e += nwaves) {
        int s = (int)src[e];
        int d = (int)dst[e];
        float norm = dinv[s] * dinv[d];
        const float* hs = H + (size_t)s * D + lane * VEC;
        float* od = out + (size_t)d * D + lane * VEC;
        float v[VEC];
#pragma unroll
        for (int c = 0; c < VEC; ++c) v[c] = hs[c];  // contiguous -> b128/b64 load
#pragma unroll
        for (int c = 0; c < VEC; ++c) atomicAdd(&od[c], v[c] * norm);
    }
}

// ---------------- batch norm ----------------

__global__ void k_zero_stats(float* __restrict__ stats) {
    stats[threadIdx.x] = 0.0f;  // 256 entries: [0..127]=sum, [128..255]=sumsq
}

__global__ void k_bn_stats(const float* __restrict__ X, int n, float* __restrict__ stats) {
    int f = threadIdx.x;  // blockDim.x == 128
    float s = 0.f, s2 = 0.f;
    for (int i = blockIdx.x; i < n; i += gridDim.x) {
        float v = X[(size_t)i * 128 + f];
        s += v;
        s2 += v * v;
    }
    atomicAdd(&stats[f], s);
    atomicAdd(&stats[128 + f], s2);
}

__global__ void k_bn_relu(float* __restrict__ X, const float* __restrict__ stats,
                          const float* __restrict__ gamma, const float* __restrict__ beta,
                          int n, int total) {
    int idx = blockIdx.x * blockDim.x + threadIdx.x;
    if (idx < total) {
        int f = idx & 127;
        float inv_n = 1.0f / (float)n;
        float mean = stats[f] * inv_n;
        float var  = stats[128 + f] * inv_n - mean * mean;  // biased, like jnp.var
        float v = (X[idx] - mean) * rsqrtf(var + BN_EPS) * gamma[f] + beta[f];
        X[idx] = v > 0.f ? v : 0.f;
    }
}

// ---------------- launcher ----------------

extern "C" void kernel_launch(void* const* d_in, const int* in_sizes, int n_in,
                              void* d_out, int out_size, void* d_ws, size_t ws_size,
                              hipStream_t stream) {
    const float*   x     = (const float*)d_in[0];
    const int64_t* edges = (const int64_t*)d_in[1];
    const float*   W1    = (const float*)d_in[2];
    const float*   b1    = (const float*)d_in[3];
    const float*   gamma = (const float*)d_in[4];
    const float*   beta  = (const float*)d_in[5];
    const float*   W2    = (const float*)d_in[6];
    const float*   b2    = (const float*)d_in[7];

    const int n = in_sizes[0] / 128;   // 100000
    const int E = in_sizes[1] / 2;     // 1600000
    const int64_t* src = edges;
    const int64_t* dst = edges + E;

    // workspace carve-up
    char* base = (char*)d_ws;
    auto align256 = [](size_t v) { return (v + 255) & ~(size_t)255; };
    float* deg  = (float*)base; base += align256((size_t)n * sizeof(float));       // deg -> dinv in place
    float* h1   = (float*)base; base += align256((size_t)n * 128 * sizeof(float)); // also reused for h2
    float* agg1 = (float*)base; base += align256((size_t)n * 128 * sizeof(float)); // also y after BN
    float* stats = (float*)base;                                                   // 256 floats

    const int T = 256;
    const int row_tiles = (n + 15) / 16;           // 6250
    const int row_quads = (row_tiles + 3) / 4;     // 1563

    // degree + dinv
    k_deg_init<<<(n + T - 1) / T, T, 0, stream>>>(deg, n);
    k_deg_count<<<2048, T, 0, stream>>>(dst, E, deg);
    k_dinv<<<(n + T - 1) / T, T, 0, stream>>>(deg, n);

    // layer 1: GEMM -> self-loop+bias -> edge scatter
    {
        int waves = row_quads * (128 / 16);
        int blocks = (waves + (T / 32) - 1) / (T / 32);
        k_gemm_wmma<128, 128><<<blocks, T, 0, stream>>>(x, W1, h1, n);
    }
    {
        int total = n * 128;
        k_init_agg<128><<<(total + T - 1) / T, T, 0, stream>>>(h1, deg, b1, agg1, total);
    }
    k_scatter<128><<<4096, T, 0, stream>>>(h1, src, dst, deg, agg1, E);

    // batch norm (training-mode) + relu, in place on agg1
    k_zero_stats<<<1, 256, 0, stream>>>(stats);
    k_bn_stats<<<1024, 128, 0, stream>>>(agg1, n, stats);
    {
        int total = n * 128;
        k_bn_relu<<<(total + T - 1) / T, T, 0, stream>>>(agg1, stats, gamma, beta, n, total);
    }

    // layer 2: GEMM (reuse h1 buffer for h2) -> init out -> edge scatter into d_out
    float* h2 = h1;
    float* out = (float*)d_out;
    {
        int waves = row_quads * (64 / 16);
        int blocks = (waves + (T / 32) - 1) / (T / 32);
        k_gemm_wmma<128, 64><<<blocks, T, 0, stream>>>(agg1, W2, h2, n);
    }
    {
        int total = n * 64;
        k_init_agg<64><<<(total + T - 1) / T, T, 0, stream>>>(h2, deg, b2, out, total);
    }
    k_scatter<64><<<4096, T, 0, stream>>>(h2, src, dst, deg, out, E);
}